// SemiMarkovModule_30459908063245
// MI455X (gfx1250) — compile-verified
//
#include <hip/hip_runtime.h>
#include <hip/hip_bf16.h>
#include <math.h>

// Problem constants (match reference)
#define Bv 16
#define Tv 2048
#define Dv 64
#define Cv 32
#define NFv 16
#define Kv 20
#define BIG_NEG (-1.0e9f)
#define NCHUNK 16
#define CHUNK  (Tv / NCHUNK)   // 128

// Workspace layout (float offsets); all 16B aligned
#define OFF_BMAT 0            // 64*32 = 2048   B-matrix [d][c] = means[c][d]*inv_var[d]
#define OFF_IVAR 2048         // 64            inv_var
#define OFF_CC   2112         // 32            per-class const: -0.5*mm[c] + log_norm
#define OFF_LEN  2144         // 20*32 = 640   Poisson length log-pmf [k][c]
#define OFF_EM   2784         // 16*2048*32 = 1048576  emission (B,T,C)
#define OFF_CS   1051360      // 16*2049*32 = 1049088  cumsum (B,T+1,C)
#define OFF_CHK  2100448      // 16*16*32 = 8192       per-chunk sums (B,NCHUNK,C)

typedef __attribute__((ext_vector_type(2))) float v2f;
typedef __attribute__((ext_vector_type(8))) float v8f;

// ---------------------------------------------------------------------------
// Prep: inv_var, Bmat, class consts, length table, init_lp (all tiny)
// ---------------------------------------------------------------------------
__global__ void k_prep(const float* __restrict__ means,
                       const float* __restrict__ stdv,
                       const float* __restrict__ plr,
                       const float* __restrict__ initl,
                       float* __restrict__ ws,
                       float* __restrict__ out_init) {
    __shared__ float s_iv[Dv];
    __shared__ float s_logsum;
    const int tid = threadIdx.x;

    if (tid < Dv) {
        float sd = stdv[tid];
        float iv = 1.0f / (sd * sd);
        s_iv[tid] = iv;
        ws[OFF_IVAR + tid] = iv;
    }
    if (tid == 0) {
        float sl = 0.0f;
        for (int d = 0; d < Dv; ++d) sl += logf(stdv[d]);
        s_logsum = sl;
    }
    __syncthreads();

    // Bmat[d][c] = means[c][d] * inv_var[d]
    for (int i = tid; i < Dv * Cv; i += blockDim.x) {
        int d = i >> 5, c = i & 31;
        ws[OFF_BMAT + i] = means[c * Dv + d] * s_iv[d];
    }
    // per-class constant: -0.5*mm[c] - sum(log std) - 0.5*D*log(2pi)
    if (tid < Cv) {
        float mm = 0.0f;
        for (int d = 0; d < Dv; ++d) {
            float m = means[tid * Dv + d];
            mm += m * m * s_iv[d];
        }
        ws[OFF_CC + tid] = -0.5f * mm - s_logsum - 0.5f * (float)Dv * 1.8378770664093453f;
    }
    // Poisson length log-pmf: (k+1)*lr - exp(lr) - lgamma(k+2)
    for (int i = tid; i < Kv * Cv; i += blockDim.x) {
        int k = i >> 5, c = i & 31;
        float lr = plr[c];
        ws[OFF_LEN + i] = (float)(k + 1) * lr - expf(lr) - lgammaf((float)(k + 2));
    }
    // init log-softmax (serial, 32 elems)
    if (tid == 0) {
        float m = -3.4e38f;
        for (int c = 0; c < Cv; ++c) m = fmaxf(m, initl[c]);
        float s = 0.0f;
        for (int c = 0; c < Cv; ++c) s += expf(initl[c] - m);
        float lse = m + logf(s);
        for (int c = 0; c < Cv; ++c) out_init[c] = initl[c] - lse;
    }
}

// ---------------------------------------------------------------------------
// Transition potentials: log_softmax((ter@P)@(tel@P)^T, axis=0) -> (32,32)
// ---------------------------------------------------------------------------
__global__ void k_trans(const float* __restrict__ proj,
                        const float* __restrict__ tel,
                        const float* __restrict__ ter,
                        float* __restrict__ outp) {
    __shared__ float sL[Cv * NFv];
    __shared__ float sR[Cv * NFv];
    __shared__ float sLg[Cv * Cv];
    __shared__ float sLse[Cv];
    const int tid = threadIdx.x;

    for (int i = tid; i < Cv * NFv; i += blockDim.x) {
        int c = i >> 4, f = i & 15;
        float al = 0.0f, ar = 0.0f;
        for (int d = 0; d < Dv; ++d) {
            float p = proj[d * NFv + f];
            al += tel[c * Dv + d] * p;
            ar += ter[c * Dv + d] * p;
        }
        sL[i] = al;
        sR[i] = ar;
    }
    __syncthreads();
    for (int i = tid; i < Cv * Cv; i += blockDim.x) {
        int r = i >> 5, cj = i & 31;  // r = target (axis 0), cj = source
        float lg = 0.0f;
        for (int f = 0; f < NFv; ++f) lg += sR[r * NFv + f] * sL[cj * NFv + f];
        sLg[i] = lg;
    }
    __syncthreads();
    if (tid < Cv) {  // column-wise logsumexp (softmax over axis=0)
        float m = -3.4e38f;
        for (int i = 0; i < Cv; ++i) m = fmaxf(m, sLg[i * Cv + tid]);
        float s = 0.0f;
        for (int i = 0; i < Cv; ++i) s += expf(sLg[i * Cv + tid] - m);
        sLse[tid] = m + logf(s);
    }
    __syncthreads();
    for (int i = tid; i < Cv * Cv; i += blockDim.x)
        outp[i] = sLg[i] - sLse[i & 31];
}

// ---------------------------------------------------------------------------
// Emission GEMM via V_WMMA_F32_16X16X4_F32:
//   em[r,c] = sum_d feat[r,d]*Bmat[d,c]  - 0.5*xx[r] + constc[c]
// Each wave: one 16-row tile, two 16x16 f32 accumulators (C=32), 16 K-steps.
// ---------------------------------------------------------------------------
__global__ void __launch_bounds__(256)
k_emission(const float* __restrict__ feat,
           const float* __restrict__ ws,
           float* __restrict__ em) {
    __shared__ float sB[Dv * Cv];   // [d][c]
    __shared__ float sIV[Dv];
    __shared__ float sCC[Cv];
    const int tid = threadIdx.x;

    for (int i = tid; i < Dv * Cv; i += 256) sB[i] = ws[OFF_BMAT + i];
    if (tid < Dv) sIV[tid] = ws[OFF_IVAR + tid];
    if (tid < Cv) sCC[tid] = ws[OFF_CC + tid];
    __syncthreads();

    const int wave = tid >> 5;
    const int lane = tid & 31;
    const int half = lane >> 4;     // 0: lanes 0-15, 1: lanes 16-31
    const int ln   = lane & 15;
    const int base = blockIdx.x * 128 + wave * 16;   // first global row of tile

    // A-matrix 16x4 f32 layout: lane<16 holds (K=0,K=1) of row=ln, lane>=16
    // holds (K=2,K=3) of row=ln  -> float2 at feat[(base+ln)*64 + 4*kk + 2*half]
    const float* fr = feat + (size_t)(base + ln) * Dv + 2 * half;

    v8f accLo = {0.f,0.f,0.f,0.f,0.f,0.f,0.f,0.f};
    v8f accHi = {0.f,0.f,0.f,0.f,0.f,0.f,0.f,0.f};
    float xxp = 0.0f;

#pragma unroll
    for (int kk = 0; kk < 16; ++kk) {
        const int d0 = 4 * kk + 2 * half;
        const v2f a = *reinterpret_cast<const v2f*>(fr + 4 * kk);
        // B-matrix 4x16: lane<16 -> (K=0,K=1) of col=ln; lane>=16 -> (K=2,K=3)
        v2f bl, bh;
        bl.x = sB[d0 * Cv + ln];        bl.y = sB[(d0 + 1) * Cv + ln];
        bh.x = sB[d0 * Cv + 16 + ln];   bh.y = sB[(d0 + 1) * Cv + 16 + ln];
        accLo = __builtin_amdgcn_wmma_f32_16x16x4_f32(false, a, false, bl,
                                                      (short)0, accLo, false, false);
        accHi = __builtin_amdgcn_wmma_f32_16x16x4_f32(false, a, false, bh,
                                                      (short)0, accHi, false, false);
        xxp += a.x * a.x * sIV[d0] + a.y * a.y * sIV[d0 + 1];
    }

    // xx for row (lane&15): combine the two K-halves across lane^16
    float xxfull = xxp + __shfl(xxp, lane ^ 16);

    const float ccLo = sCC[ln];
    const float ccHi = sCC[16 + ln];
#pragma unroll
    for (int j = 0; j < 8; ++j) {
        const int rowsel = j + 8 * half;            // M of this VGPR for this lane-half
        const float rt = __shfl(xxfull, rowsel);    // xx of that row
        const int r = base + rowsel;
        em[r * Cv + ln]      = accLo[j] - 0.5f * rt + ccLo;
        em[r * Cv + 16 + ln] = accHi[j] - 0.5f * rt + ccHi;
    }
}

// ---------------------------------------------------------------------------
// Blocked scan, phase A: per-(b,chunk,c) partial sums over CHUNK timesteps.
// 8192 threads; lanes vary over c -> fully coalesced 128B accesses.
// ---------------------------------------------------------------------------
__global__ void k_chunksum(const float* __restrict__ em, float* __restrict__ chk) {
    const int g = blockIdx.x * blockDim.x + threadIdx.x;  // 0..8191
    const int c = g & 31;
    const int chunk = (g >> 5) & (NCHUNK - 1);
    const int b = g >> 9;
    const float* e = em + ((size_t)(b * Tv + chunk * CHUNK)) * Cv + c;
    float s = 0.0f;
    for (int t = 0; t < CHUNK; ++t) s += e[(size_t)t * Cv];
    chk[(b * NCHUNK + chunk) * Cv + c] = s;
}

// ---------------------------------------------------------------------------
// Blocked scan, phase B: base = prefix of chunk sums, then serial cumsum over
// this thread's 128-step chunk. Dependent chain 128 (vs 2048), 256 waves.
// ---------------------------------------------------------------------------
__global__ void k_cumsum(const float* __restrict__ em,
                         const float* __restrict__ chk,
                         float* __restrict__ cs) {
    const int g = blockIdx.x * blockDim.x + threadIdx.x;  // 0..8191
    const int c = g & 31;
    const int chunk = (g >> 5) & (NCHUNK - 1);
    const int b = g >> 9;

    float acc = 0.0f;
    for (int j = 0; j < chunk; ++j) acc += chk[(b * NCHUNK + j) * Cv + c];

    if (chunk == 0) cs[(size_t)(b * (Tv + 1)) * Cv + c] = 0.0f;

    const float* e = em + ((size_t)(b * Tv + chunk * CHUNK)) * Cv + c;
    float* o = cs + ((size_t)(b * (Tv + 1) + chunk * CHUNK + 1)) * Cv + c;
    for (int t = 0; t < CHUNK; ++t) {
        acc += e[(size_t)t * Cv];
        o[(size_t)t * Cv] = acc;
    }
}

// ---------------------------------------------------------------------------
// Span writer: one float4 (4 classes) per thread; flat out offset = 4*threadId
// out[b,k,t,c] = (t>=k ? cs[b,t+1,c]-cs[b,t-k,c] : BIG_NEG) + len_lp[k,c]
// ---------------------------------------------------------------------------
__global__ void __launch_bounds__(256)
k_span(const float* __restrict__ cs,
       const float* __restrict__ len,
       float* __restrict__ out) {
    const unsigned i = blockIdx.x * 256u + threadIdx.x;  // 0 .. 5242879
    const int c4 = (i & 7) << 2;
    const int t  = (i >> 3) & (Tv - 1);
    const unsigned q = i >> 14;       // b*K + k  (T*C/4 = 16384)
    const int k = (int)(q % Kv);
    const int b = (int)(q / Kv);

    const int st = t - k;
    const bool valid = (st >= 0);
    const int stc = valid ? st : 0;

    const float4 e  = *reinterpret_cast<const float4*>(
        cs + (size_t)(b * (Tv + 1) + t + 1) * Cv + c4);
    const float4 s0 = *reinterpret_cast<const float4*>(
        cs + (size_t)(b * (Tv + 1) + stc) * Cv + c4);
    const float4 lp = *reinterpret_cast<const float4*>(len + k * Cv + c4);

    float4 r;
    r.x = (valid ? e.x - s0.x : BIG_NEG) + lp.x;
    r.y = (valid ? e.y - s0.y : BIG_NEG) + lp.y;
    r.z = (valid ? e.z - s0.z : BIG_NEG) + lp.z;
    r.w = (valid ? e.w - s0.w : BIG_NEG) + lp.w;
    *reinterpret_cast<float4*>(out + (size_t)i * 4) = r;
}

// ---------------------------------------------------------------------------
extern "C" void kernel_launch(void* const* d_in, const int* in_sizes, int n_in,
                              void* d_out, int out_size, void* d_ws, size_t ws_size,
                              hipStream_t stream) {
    const float* feat  = (const float*)d_in[0];   // (B,T,D)
    const float* proj  = (const float*)d_in[1];   // (D,NF)
    const float* tel   = (const float*)d_in[2];   // (C,D)
    const float* ter   = (const float*)d_in[3];   // (C,D)
    const float* means = (const float*)d_in[4];   // (C,D)
    const float* stdv  = (const float*)d_in[5];   // (D,)
    const float* plr   = (const float*)d_in[6];   // (C,)
    const float* initl = (const float*)d_in[7];   // (C,)
    float* out = (float*)d_out;
    float* ws  = (float*)d_ws;

    const size_t span_elems = (size_t)Bv * Kv * Tv * Cv;   // 20,971,520
    float* out_trans = out + span_elems;                   // 1024
    float* out_init  = out_trans + (size_t)Cv * Cv;        // 32

    k_prep<<<1, 256, 0, stream>>>(means, stdv, plr, initl, ws, out_init);
    k_trans<<<1, 256, 0, stream>>>(proj, tel, ter, out_trans);
    k_emission<<<(Bv * Tv) / 128, 256, 0, stream>>>(feat, ws, ws + OFF_EM);
    k_chunksum<<<32, 256, 0, stream>>>(ws + OFF_EM, ws + OFF_CHK);
    k_cumsum<<<32, 256, 0, stream>>>(ws + OFF_EM, ws + OFF_CHK, ws + OFF_CS);
    k_span<<<(unsigned)(span_elems / 4 / 256), 256, 0, stream>>>(
        ws + OFF_CS, ws + OFF_LEN, out);
}